// NettackSurrogate_48009144435494
// MI455X (gfx1250) — compile-verified
//
#include <hip/hip_runtime.h>
#include <hip/hip_bf16.h>
#include <stdint.h>

typedef __attribute__((ext_vector_type(2))) float v2f;
typedef __attribute__((ext_vector_type(8))) float v8f;

#define IN_CH 512
#define OUT_CH 64

// ---------------- helpers ----------------

__global__ void zero_f32_kernel(float* __restrict__ p, int n) {
    int i = blockIdx.x * blockDim.x + threadIdx.x;
    if (i < n) p[i] = 0.0f;
}

// deg[row] += 1 for non-self-loop edges
__global__ void deg_count_kernel(const long long* __restrict__ rows,
                                 const long long* __restrict__ cols,
                                 float* __restrict__ deg, int n_edges) {
    int e = blockIdx.x * blockDim.x + threadIdx.x;
    if (e < n_edges) {
        int r = (int)rows[e];
        int c = (int)cols[e];
        if (r != c) atomicAdd(&deg[r], 1.0f);
    }
}

// deg -> rsqrt(deg + 1)   (in place; +1 is the self loop)
__global__ void dinv_kernel(float* __restrict__ deg, int n) {
    int i = blockIdx.x * blockDim.x + threadIdx.x;
    if (i < n) deg[i] = __frsqrt_rn(deg[i] + 1.0f);
}

// ---------------- GEMM: h = x(n,512) @ W(512,64) via V_WMMA_F32_16X16X4_F32 ----
// Block = 256 threads = 8 waves. W (512x64 f32 = 128KB) is staged once into
// CDNA5's 320KB LDS; B fragments then come from ds_load, A from one b64/lane.
// Wave w: row_tile = blockIdx.x*2 + (w>>2), col_tile = w&3.
__global__ void __launch_bounds__(256)
gemm_wmma_kernel(const float* __restrict__ x, const float* __restrict__ W,
                 float* __restrict__ h, int n_nodes) {
    extern __shared__ float sW[];   // IN_CH*OUT_CH floats = 128KB

    // cooperative stage of W into LDS (32 x b128 per thread)
    {
        const float4* __restrict__ Wv = (const float4*)W;
        float4* sWv = (float4*)sW;
        for (int i = threadIdx.x; i < (IN_CH * OUT_CH) / 4; i += 256)
            sWv[i] = Wv[i];
    }
    __syncthreads();

    const int lane = threadIdx.x & 31;
    const int wave = threadIdx.x >> 5;
    const int row_tile = blockIdx.x * 2 + (wave >> 2);
    const int col0 = (wave & 3) * 16;
    const int m = lane & 15;
    const int khalf = (lane >> 4) * 2;   // 0 for lanes 0-15, 2 for lanes 16-31

    const int row = row_tile * 16 + m;
    const int row_clamped = row < n_nodes ? row : (n_nodes - 1);

    const float* __restrict__ arow = x + (size_t)row_clamped * IN_CH;
    const float* bcol = sW + col0 + m;   // + k*64 per step (LDS)

    v8f acc = {};
#pragma unroll 4
    for (int k = 0; k < IN_CH; k += 4) {
        // A fragment: row m, K = {k+khalf, k+khalf+1}; 8B-aligned float2 load
        v2f a = *(const v2f*)(arow + k + khalf);
        // B fragment from LDS: N = col0+m, K rows {k+khalf, k+khalf+1}
        const float* bp = bcol + (k + khalf) * OUT_CH;
        v2f b;
        b.x = bp[0];
        b.y = bp[OUT_CH];
        acc = __builtin_amdgcn_wmma_f32_16x16x4_f32(
            /*neg_a=*/false, a, /*neg_b=*/false, b,
            /*c_mod=*/(short)0, acc, /*reuse_a=*/false, /*reuse_b=*/false);
    }

    // C/D layout: VGPR v -> M = v + (lane<16 ? 0 : 8), N = lane&15
    const int mbase = row_tile * 16 + (lane >> 4) * 8;
    const int n = col0 + m;
    float* __restrict__ dst = h + (size_t)mbase * OUT_CH + n;
    if (row_tile * 16 + 16 <= n_nodes) {
        // full tile: branch-free strided stores
#pragma unroll
        for (int v = 0; v < 8; v++) dst[(size_t)v * OUT_CH] = acc[v];
    } else {
#pragma unroll
        for (int v = 0; v < 8; v++)
            if (mbase + v < n_nodes) dst[(size_t)v * OUT_CH] = acc[v];
    }
}

// hout[i, c] = dinv[i]^2 * hin[i, c]   (initializes the segment-sum target)
__global__ void self_scale_kernel(const float* __restrict__ dinv,
                                  const float* __restrict__ hin,
                                  float* __restrict__ hout, long long total) {
    long long i = (long long)blockIdx.x * blockDim.x + threadIdx.x;
    if (i < total) {
        int node = (int)(i >> 6);        // OUT_CH == 64
        float d = dinv[node];
        hout[i] = d * d * hin[i];
    }
}

// One wave32 per edge; each lane owns a float2 of the 64 channels.
// Edge indices + dinv are wave-uniform -> force scalar loads via readfirstlane.
// hout[row, :] += dinv[row]*dinv[col] * hin[col, :]
__global__ void __launch_bounds__(256)
scatter_kernel(const long long* __restrict__ rows,
               const long long* __restrict__ cols,
               const float* __restrict__ dinv,
               const float* __restrict__ hin,
               float* __restrict__ hout, int n_edges) {
    int e = __builtin_amdgcn_readfirstlane(blockIdx.x * 8 + (threadIdx.x >> 5));
    int lane = threadIdx.x & 31;
    if (e >= n_edges) return;
    int r = __builtin_amdgcn_readfirstlane((int)rows[e]);
    int c = __builtin_amdgcn_readfirstlane((int)cols[e]);
    if (r == c) return;                  // remove_self_loops
    float norm = dinv[r] * dinv[c];      // scalar operands
    v2f v = *(const v2f*)(hin + (size_t)c * OUT_CH + lane * 2);
    float* dst = hout + (size_t)r * OUT_CH + lane * 2;
    atomicAdd(dst,     norm * v.x);
    atomicAdd(dst + 1, norm * v.y);
}

// ---------------- launch ----------------

extern "C" void kernel_launch(void* const* d_in, const int* in_sizes, int n_in,
                              void* d_out, int out_size, void* d_ws, size_t ws_size,
                              hipStream_t stream) {
    const long long* edge = (const long long*)d_in[0];  // int64 (2, E)
    const float*     x    = (const float*)d_in[1];      // (N, 512)
    const float*     W    = (const float*)d_in[2];      // (512, 64)
    float*           out  = (float*)d_out;              // (N, 64)

    const int n_edges = in_sizes[0] / 2;
    const int n_nodes = in_sizes[1] / IN_CH;
    const long long total = (long long)n_nodes * OUT_CH;

    const long long* rows = edge;
    const long long* cols = edge + n_edges;

    // workspace layout: [dinv | h0 | h1]
    float* dinv = (float*)d_ws;
    size_t off  = (((size_t)n_nodes * sizeof(float)) + 255) & ~(size_t)255;
    float* h0   = (float*)((char*)d_ws + off);
    float* h1   = h0 + total;

    const int B = 256;

    // 1) degree -> dinv
    zero_f32_kernel<<<(n_nodes + B - 1) / B, B, 0, stream>>>(dinv, n_nodes);
    deg_count_kernel<<<(n_edges + B - 1) / B, B, 0, stream>>>(rows, cols, dinv, n_edges);
    dinv_kernel<<<(n_nodes + B - 1) / B, B, 0, stream>>>(dinv, n_nodes);

    // 2) h0 = x @ W  (WMMA, W staged in 128KB LDS)
    int row_tiles = (n_nodes + 15) / 16;
    int gemm_blocks = (row_tiles + 1) / 2;   // 2 row tiles (8 waves) per block
    gemm_wmma_kernel<<<gemm_blocks, B, (size_t)IN_CH * OUT_CH * sizeof(float),
                       stream>>>(x, W, h0, n_nodes);

    // 3) hop 1: h1 = self_norm*h0 + segment_sum(norm*h0[col], row)
    self_scale_kernel<<<(int)((total + B - 1) / B), B, 0, stream>>>(dinv, h0, h1, total);
    scatter_kernel<<<(n_edges + 7) / 8, B, 0, stream>>>(rows, cols, dinv, h0, h1, n_edges);

    // 4) hop 2: out = self_norm*h1 + segment_sum(norm*h1[col], row)
    self_scale_kernel<<<(int)((total + B - 1) / B), B, 0, stream>>>(dinv, h1, out, total);
    scatter_kernel<<<(n_edges + 7) / 8, B, 0, stream>>>(rows, cols, dinv, h1, out, n_edges);
}